// PolicyNet_41807211659790
// MI455X (gfx1250) — compile-verified
//
#include <hip/hip_runtime.h>

typedef float v2f __attribute__((ext_vector_type(2)));
typedef float v4f __attribute__((ext_vector_type(4)));
typedef float v8f __attribute__((ext_vector_type(8)));

__device__ __forceinline__ float fast_rcp(float x) { return __builtin_amdgcn_rcpf(x); }
__device__ __forceinline__ float sigmoid1(float x) { return fast_rcp(1.0f + __expf(-x)); }
__device__ __forceinline__ float tanh1(float x) {
    // stable: tanh(x) = sign(x) * (1 - e) / (1 + e), e = exp(-2|x|)
    float e = __expf(-2.0f * fabsf(x));
    float t = (1.0f - e) * fast_rcp(1.0f + e);
    return copysignf(t, x);
}
__device__ __forceinline__ v4f sigmoid4(v4f x) {
    return (v4f){sigmoid1(x.x), sigmoid1(x.y), sigmoid1(x.z), sigmoid1(x.w)};
}
__device__ __forceinline__ v4f tanh4(v4f x) {
    return (v4f){tanh1(x.x), tanh1(x.y), tanh1(x.z), tanh1(x.w)};
}

// One block = 256 threads = 8 waves; each wave handles 32 rows (two 16x16x4 WMMA tiles).
__global__ __launch_bounds__(256) void policy_lstm_kernel(
    const float* __restrict__ inp,    // [B,12] : h(0:4) c(4:8) obs(8:12)
    const float* __restrict__ w_ih,   // [16,4]
    const float* __restrict__ w_hh,   // [16,4]
    const float* __restrict__ b_ih,   // [16]
    const float* __restrict__ b_hh,   // [16]
    const float* __restrict__ w_lin,  // [2,4]
    const float* __restrict__ b_lin,  // [2]
    float* __restrict__ out,          // [2B] dist ++ [8B] hc
    int Btot)
{
    __shared__ float lds[8 * 32 * 16];   // 8 waves x 32 rows x 16 gates = 16 KB

    const int tid  = threadIdx.x;
    const int lane = tid & 31;
    const int wave = tid >> 5;
    const int half = lane >> 4;   // 0: K={0,1} / rows M 0..7 ; 1: K={2,3} / rows M 8..15
    const int n    = lane & 15;   // gate column / A-matrix row within tile

    const int blockRow0 = blockIdx.x * 256;
    const int waveRow0  = blockRow0 + wave * 32;

    // ---- B operands (4x16 weight matrices, K = vgpr + 2*half, N = lane%16) ----
    // w_ihT[k][n] = w_ih[n][k]; rows of w_ih are contiguous 4 floats -> b64 loads.
    v2f bwih = *(const v2f*)(w_ih + n * 4 + 2 * half);
    v2f bwhh = *(const v2f*)(w_hh + n * 4 + 2 * half);
    const float cb = b_ih[n] + b_hh[n];   // bias broadcast down the M dimension

    // ---- gate GEMM via chained V_WMMA_F32_16X16X4_F32 ----
#pragma unroll
    for (int t = 0; t < 2; ++t) {
        int rowm = waveRow0 + t * 16 + n;          // A-matrix row (M = lane%16)
        rowm = min(rowm, Btot - 1);                // keep EXEC all-ones, avoid OOB
        const float* rp = inp + (size_t)rowm * 12;
        v2f ah = *(const v2f*)(rp + 2 * half);       // h   cols {0,1}/{2,3}
        v2f ao = *(const v2f*)(rp + 8 + 2 * half);   // obs cols {8,9}/{10,11}

        v8f acc = {cb, cb, cb, cb, cb, cb, cb, cb};
        acc = __builtin_amdgcn_wmma_f32_16x16x4_f32(false, ah, false, bwhh,
                                                    (short)0, acc, false, false);
        acc = __builtin_amdgcn_wmma_f32_16x16x4_f32(false, ao, false, bwih,
                                                    (short)0, acc, false, false);

        // D layout: vgpr v -> row (v + 8*half), col n. Scatter tile to LDS.
        float* lp = &lds[wave * 512 + (t * 16 + half * 8) * 16 + n];
#pragma unroll
        for (int v = 0; v < 8; ++v) lp[v * 16] = acc[v];
    }

    __syncthreads();

    // ---- per-row elementwise stage: lane owns one full row ----
    const int R  = blockRow0 + tid;                 // global row this lane finishes
    const int Rc = min(R, Btot - 1);

    const float* lrow = &lds[tid * 16];
    v4f Gi = *(const v4f*)(lrow + 0);
    v4f Gf = *(const v4f*)(lrow + 4);
    v4f Gg = *(const v4f*)(lrow + 8);
    v4f Go = *(const v4f*)(lrow + 12);

    v4f cOld = *(const v4f*)(inp + (size_t)Rc * 12 + 4);  // 16B-aligned (48B rows)

    v4f i_ = sigmoid4(Gi);
    v4f f_ = sigmoid4(Gf);
    v4f g_ = tanh4(Gg);
    v4f o_ = sigmoid4(Go);

    v4f cNew = f_ * cOld + i_ * g_;
    v4f hNew = o_ * tanh4(cNew);

    // linear head + 2-way softmax
    v4f wl0 = *(const v4f*)(w_lin);
    v4f wl1 = *(const v4f*)(w_lin + 4);
    float l0 = hNew.x * wl0.x + hNew.y * wl0.y + hNew.z * wl0.z + hNew.w * wl0.w + b_lin[0];
    float l1 = hNew.x * wl1.x + hNew.y * wl1.y + hNew.z * wl1.z + hNew.w * wl1.w + b_lin[1];
    float p0 = fast_rcp(1.0f + __expf(l1 - l0));
    float p1 = 1.0f - p0;

    if (R < Btot) {
        v2f pd = {p0, p1};
        *(v2f*)(out + (size_t)R * 2) = pd;                       // action_dist
        float* hcp = out + (size_t)Btot * 2 + (size_t)R * 8;     // hc = [h_new, c_new]
        *(v4f*)(hcp)     = hNew;
        *(v4f*)(hcp + 4) = cNew;
    }
}

extern "C" void kernel_launch(void* const* d_in, const int* in_sizes, int n_in,
                              void* d_out, int out_size, void* d_ws, size_t ws_size,
                              hipStream_t stream) {
    const float* inp   = (const float*)d_in[0];
    const float* w_ih  = (const float*)d_in[1];
    const float* w_hh  = (const float*)d_in[2];
    const float* b_ih  = (const float*)d_in[3];
    const float* b_hh  = (const float*)d_in[4];
    const float* w_lin = (const float*)d_in[5];
    const float* b_lin = (const float*)d_in[6];
    float* out = (float*)d_out;

    const int B = in_sizes[0] / 12;        // 4194304 in the reference
    const int grid = (B + 255) / 256;      // B is divisible by 256
    policy_lstm_kernel<<<grid, 256, 0, stream>>>(inp, w_ih, w_hh, b_ih, b_hh,
                                                 w_lin, b_lin, out, B);
}